// FCOSloss_62740882260765
// MI455X (gfx1250) — compile-verified
//
#include <hip/hip_runtime.h>
#include <hip/hip_bf16.h>

// Problem constants (from reference)
#define BB   16
#define CC   80
#define PP   21824          // 128^2 + 64^2 + 32^2 + 16^2 + 8^2
#define CP   (CC * PP)      // 1,745,920 elements per image
#define N4   (CP / 4)       // 436,480 float4s per image (divisible by 256)

typedef __attribute__((ext_vector_type(2))) float v2f;
typedef __attribute__((ext_vector_type(4))) float v4f;
typedef __attribute__((ext_vector_type(8))) float v8f;
typedef __attribute__((ext_vector_type(4))) int   v4i;

// D = A(16x4) * ones(4x16) + C : accumulates all 64 f32 elements of A into C
// (each row-sum replicated across 16 columns -> final total must be * 1/16).
#if defined(__has_builtin) && __has_builtin(__builtin_amdgcn_wmma_f32_16x16x4_f32)
#define WMMA_ACC(c, a, b)                                                      \
  (c) = __builtin_amdgcn_wmma_f32_16x16x4_f32(false, (a), false, (b),          \
                                              (short)0, (c), false, false)
#else
#define WMMA_ACC(c, a, b)                                                      \
  asm volatile("v_wmma_f32_16x16x4_f32 %0, %1, %2, %0"                         \
               : "+v"(c) : "v"(a), "v"(b))
#endif

__device__ __forceinline__ float wave_reduce_sum(float v) {
  v += __shfl_xor(v, 16, 32);
  v += __shfl_xor(v, 8, 32);
  v += __shfl_xor(v, 4, 32);
  v += __shfl_xor(v, 2, 32);
  v += __shfl_xor(v, 1, 32);
  return v;
}

// ws layout (floats): [0..15] loss_conf, [16..31] loss_l, [32..47] loss_center,
//                     [48..63] poses
__global__ void zero_ws_kernel(float* ws) {
  if (threadIdx.x < 64) ws[threadIdx.x] = 0.0f;
}

// focal loss term for one element
__device__ __forceinline__ float focal(float conf, bool pos) {
  float p = fminf(fmaxf(conf, 1e-8f), 0.99999999f);
  float om = 1.0f - p;
  float arg = pos ? p : om;               // one log per element
  float w   = pos ? (0.25f * om * om) : (0.75f * p * p);
  return -w * __logf(arg);
}

// Kernel 1: focal loss over confs(B,C,P), per-image sums via WMMA accumulate.
__global__ __launch_bounds__(256) void focal_kernel(
    const float* __restrict__ confs, const int* __restrict__ tag_class,
    float* __restrict__ ws) {
  const int b = blockIdx.y;
  const v4f* __restrict__ cf =
      reinterpret_cast<const v4f*>(confs + (size_t)b * CP);
  const v4i* __restrict__ tg =
      reinterpret_cast<const v4i*>(tag_class + (size_t)b * PP);

  v8f acc = {};                 // WMMA accumulator (16x16 f32)
  v2f ones; ones.x = 1.0f; ones.y = 1.0f;   // B = all ones (4x16)

  const int stride = gridDim.x * blockDim.x;        // in float4 units
  const int base   = blockIdx.x * blockDim.x;       // block-uniform
  // N4 % 256 == 0 -> the residual (N4 - base) mod stride is a multiple of 256,
  // so this trip count is identical for every lane in the block. A counted
  // scalar loop keeps the branch on SALU and guarantees EXEC == all-ones at
  // every WMMA (ISA requirement), with no per-lane compare in the hot loop.
  const int niter = (N4 - base + stride - 1) / stride;

  int idx = base + threadIdx.x;
  #pragma unroll 2
  for (int k = 0; k < niter; ++k, idx += stride) {
    const int flat = idx * 4;           // element index within (C,P)
    const int c    = flat / PP;         // channel (P % 4 == 0 -> uniform in f4)
    const int p    = flat - c * PP;     // pixel index, p % 4 == 0

    // confs is a 112 MB single-use stream: non-temporal so it doesn't evict
    // the small tag_class rows (87 KB/image) that are re-read 80x from L2.
    v4f v  = __builtin_nontemporal_load(cf + idx);      // b128, TH=NT
    v4i tc = tg[p >> 2];                                // b128, L2-resident

    v2f a0, a1;
    a0.x = focal(v.x, tc.x == c);
    a0.y = focal(v.y, tc.y == c);
    a1.x = focal(v.z, tc.z == c);
    a1.y = focal(v.w, tc.w == c);

    WMMA_ACC(acc, a0, ones);            // v_wmma_f32_16x16x4_f32
    WMMA_ACC(acc, a1, ones);
  }

  // Sum this lane's 8 accumulator slots, then reduce across wave & block.
  float s = acc[0] + acc[1] + acc[2] + acc[3] +
            acc[4] + acc[5] + acc[6] + acc[7];
  s = wave_reduce_sum(s);

  __shared__ float red[8];
  const int lane = threadIdx.x & 31;
  const int wid  = threadIdx.x >> 5;
  if (lane == 0) red[wid] = s;
  __syncthreads();
  if (threadIdx.x == 0) {
    float t = 0.0f;
    #pragma unroll
    for (int i = 0; i < 8; ++i) t += red[i];
    atomicAdd(ws + b, t * 0.0625f);     // undo 16x column replication (exact)
  }
}

// Kernel 2: per-(b,p) IoU log-loss + centerness BCE + pos count.
__global__ __launch_bounds__(256) void box_kernel(
    const float* __restrict__ locs,      // (B,4,P)
    const float* __restrict__ centers,   // (B,P)
    const float* __restrict__ tag_box,   // (B,P,4)
    const float* __restrict__ center_t,  // (B,P)
    const float* __restrict__ pixel_xy,  // (P,2)
    const int* __restrict__ status,      // (B,P)
    float* __restrict__ ws) {
  const int b = blockIdx.y;
  const int p = blockIdx.x * blockDim.x + threadIdx.x;

  float l_contrib = 0.0f, c_contrib = 0.0f, pose = 0.0f;
  if (p < PP) {
    const int bp = b * PP + p;
    const float x = pixel_xy[2 * p + 0];
    const float y = pixel_xy[2 * p + 1];

    const float* lb = locs + (size_t)b * 4 * PP;
    const float pl = x - lb[p];
    const float pt = y - lb[PP + p];
    const float pr = x + lb[2 * PP + p];
    const float pb = y + lb[3 * PP + p];

    const v4f t4 = reinterpret_cast<const v4f*>(tag_box)[bp];
    const float tl = t4.x, tt = t4.y, tr = t4.z, tb = t4.w;

    const float s1 = (tb - tt + 1.0f) * (tr - tl + 1.0f);
    const float s2 = (pb - pt + 1.0f) * (pr - pl + 1.0f);
    const float cl = fmaxf(tl, pl), cr = fminf(tr, pr);
    const float ct = fmaxf(tt, pt), cb = fminf(tb, pb);
    const float sc = (cr - cl + 1.0f) * (cb - ct + 1.0f);
    const float un = s1 + s2 - sc;
    const bool valid = (cl < cr) && (ct < cb) && (sc > 0.0f) && (un > 0.0f);
    const float iou_l = valid ? -__logf(sc / un) : 0.0f;

    const float cen = centers[bp];
    const float ctt = center_t[bp];
    const float bce = -(ctt * fmaxf(__logf(cen), -100.0f) +
                        (1.0f - ctt) * fmaxf(log1pf(-cen), -100.0f));

    const float pos = (float)status[bp];
    l_contrib = iou_l * pos;
    c_contrib = bce * pos;
    pose      = pos;
  }

  l_contrib = wave_reduce_sum(l_contrib);
  c_contrib = wave_reduce_sum(c_contrib);
  pose      = wave_reduce_sum(pose);

  __shared__ float red[3][8];
  const int lane = threadIdx.x & 31;
  const int wid  = threadIdx.x >> 5;
  if (lane == 0) { red[0][wid] = l_contrib; red[1][wid] = c_contrib; red[2][wid] = pose; }
  __syncthreads();
  if (threadIdx.x == 0) {
    float tl_ = 0.0f, tc_ = 0.0f, tp_ = 0.0f;
    #pragma unroll
    for (int i = 0; i < 8; ++i) { tl_ += red[0][i]; tc_ += red[1][i]; tp_ += red[2][i]; }
    atomicAdd(ws + 16 + b, tl_);
    atomicAdd(ws + 32 + b, tc_);
    atomicAdd(ws + 48 + b, tp_);
  }
}

// Kernel 3: combine per image, mean over B.
__global__ void finalize_kernel(const float* __restrict__ ws,
                                float* __restrict__ out) {
  const int t = threadIdx.x;
  float per = 0.0f;
  if (t < BB) {
    const float conf  = ws[t];
    const float l     = ws[16 + t];
    const float cen   = ws[32 + t];
    const float poses = ws[48 + t];
    per = (poses > 0.0f) ? (cen + (conf + l) / fmaxf(poses, 1.0f))
                         : (cen + conf + l);
  }
  per = wave_reduce_sum(per);
  if (t == 0) out[0] = per * (1.0f / (float)BB);
}

extern "C" void kernel_launch(void* const* d_in, const int* in_sizes, int n_in,
                              void* d_out, int out_size, void* d_ws,
                              size_t ws_size, hipStream_t stream) {
  (void)in_sizes; (void)n_in; (void)out_size; (void)ws_size;
  const float* confs    = (const float*)d_in[0];
  const float* locs     = (const float*)d_in[1];
  const float* centers  = (const float*)d_in[2];
  const float* tag_box  = (const float*)d_in[3];
  const float* center_t = (const float*)d_in[4];
  const float* pixel_xy = (const float*)d_in[5];
  const int*   tag_cls  = (const int*)d_in[6];
  const int*   status   = (const int*)d_in[7];
  float* out = (float*)d_out;
  float* ws  = (float*)d_ws;

  zero_ws_kernel<<<1, 64, 0, stream>>>(ws);

  // Heavy streaming pass over confs: 2048 blocks x 256 threads (wave32: 8
  // waves/block), enough in-flight waves to saturate 23.3 TB/s.
  focal_kernel<<<dim3(128, BB), 256, 0, stream>>>(confs, tag_cls, ws);

  // Light pass over (B,P).
  const int blocks_p = (PP + 255) / 256;   // 86
  box_kernel<<<dim3(blocks_p, BB), 256, 0, stream>>>(
      locs, centers, tag_box, center_t, pixel_xy, status, ws);

  finalize_kernel<<<1, 32, 0, stream>>>(ws, out);
}